// GranularityAwarenessSensor_32650341384789
// MI455X (gfx1250) — compile-verified
//
#include <hip/hip_runtime.h>
#include <hip/hip_bf16.h>

// Problem constants (from reference)
#define B_  8
#define H_  12
#define N_  1024
#define D_  768
#define K_  45
#define NP1 1025
#define MPAD 48   // K_ padded to 3 WMMA M-tiles

typedef __attribute__((ext_vector_type(2))) float v2f;
typedef __attribute__((ext_vector_type(8))) float v8f;

// ---------------------------------------------------------------------------
// Kernel 1: A_cls[b,n] = mean_h attention[b,h,0,1+n]
// ---------------------------------------------------------------------------
__global__ void k_acls(const float* __restrict__ attn, float* __restrict__ acls) {
    int i = blockIdx.x * blockDim.x + threadIdx.x;
    if (i >= B_ * N_) return;
    int b = i >> 10, n = i & (N_ - 1);
    const size_t HS = (size_t)NP1 * NP1;                 // one head matrix
    const float* p = attn + (size_t)b * H_ * HS + 1 + n; // row 0, col 1+n
    float s = 0.f;
    #pragma unroll
    for (int h = 0; h < H_; ++h) s += p[(size_t)h * HS];
    acls[i] = s / 12.0f;
}

// ---------------------------------------------------------------------------
// argmax reduce helper semantics: larger value wins; equal value -> smaller idx
// ---------------------------------------------------------------------------
__device__ __forceinline__ void red_pair(float& v, int& id) {
    #pragma unroll
    for (int off = 16; off > 0; off >>= 1) {
        float ov = __shfl_down(v, off, 32);
        int   oi = __shfl_down(id, off, 32);
        if (ov > v || (ov == v && oi < id)) { v = ov; id = oi; }
    }
}

// ---------------------------------------------------------------------------
// Kernel 2: iterative top-45 per batch (matches lax.top_k ordering)
// one block of 1024 threads per batch
// ---------------------------------------------------------------------------
__global__ void k_topk(const float* __restrict__ acls, int* __restrict__ topk) {
    const int b = blockIdx.x;
    const int t = threadIdx.x;
    __shared__ float vals[N_];
    __shared__ float rv[32];
    __shared__ int   ri[32];
    __shared__ int   bestIdx;
    vals[t] = acls[b * N_ + t];
    __syncthreads();
    for (int k = 0; k < K_; ++k) {
        float v = vals[t]; int id = t;
        red_pair(v, id);
        if ((t & 31) == 0) { rv[t >> 5] = v; ri[t >> 5] = id; }
        __syncthreads();
        if (t < 32) {
            float v2 = rv[t]; int i2 = ri[t];
            red_pair(v2, i2);
            if (t == 0) { bestIdx = i2; topk[b * K_ + k] = i2; }
        }
        __syncthreads();
        if (t == bestIdx) vals[t] = -3.0e38f;
        __syncthreads();
    }
}

// ---------------------------------------------------------------------------
// Kernel 3: sims[b, 0:48, 0:1024] = Q (gathered rows) x F^T  via
// V_WMMA_F32_16X16X4_F32.  grid=(8,3,4), block=128 (4 waves).
// Each wave: one 16-row M tile x four 16-col N tiles, loop K=768 in steps of 4.
// ---------------------------------------------------------------------------
__global__ void k_sims_wmma(const float* __restrict__ feats,
                            const int* __restrict__ topk,
                            float* __restrict__ sims) {
    const int b    = blockIdx.x;
    const int mt   = blockIdx.y;
    const int wave = threadIdx.x >> 5;
    const int lane = threadIdx.x & 31;
    const int half = lane >> 4;      // K sub-split within fragment
    const int l16  = lane & 15;
    const int nt0  = (blockIdx.z * 4 + wave) * 4;  // first of 4 N tiles

    // A fragment source row: gathered top-k feature row (padded rows reuse 44)
    int m = mt * 16 + l16;
    int qrow = topk[b * K_ + (m < K_ ? m : (K_ - 1))];
    const float* aptr = feats + ((size_t)b * N_ + qrow) * D_ + 2 * half;
    // B fragment source rows: columns n = nt*16 + l16, element k at F[n][d+k]
    const float* bptr = feats + ((size_t)b * N_ + (nt0 * 16 + l16)) * D_ + 2 * half;

    v8f c0 = {}, c1 = {}, c2 = {}, c3 = {};
    for (int d = 0; d < D_; d += 4) {
        v2f a  = *(const v2f*)(aptr + d);
        v2f b0 = *(const v2f*)(bptr + d);
        v2f b1 = *(const v2f*)(bptr + 16 * D_ + d);
        v2f b2 = *(const v2f*)(bptr + 32 * D_ + d);
        v2f b3 = *(const v2f*)(bptr + 48 * D_ + d);
        c0 = __builtin_amdgcn_wmma_f32_16x16x4_f32(false, a, false, b0, (short)0, c0, false, false);
        c1 = __builtin_amdgcn_wmma_f32_16x16x4_f32(false, a, false, b1, (short)0, c1, false, false);
        c2 = __builtin_amdgcn_wmma_f32_16x16x4_f32(false, a, false, b2, (short)0, c2, false, false);
        c3 = __builtin_amdgcn_wmma_f32_16x16x4_f32(false, a, false, b3, (short)0, c3, false, false);
    }

    // C/D layout: vgpr r, lane half 0 -> M=r, half 1 -> M=r+8 ; N = l16
    float* out = sims + ((size_t)b * MPAD + mt * 16 + 8 * half) * N_ + nt0 * 16 + l16;
    #pragma unroll
    for (int r = 0; r < 8; ++r) {
        out[(size_t)r * N_ +  0] = c0[r];
        out[(size_t)r * N_ + 16] = c1[r];
        out[(size_t)r * N_ + 32] = c2[r];
        out[(size_t)r * N_ + 48] = c3[r];
    }
}

// ---------------------------------------------------------------------------
// Kernel 4: sequential greedy diversity selection + final mask.
// one block of 1024 threads per batch; mask kept in LDS.
// Replicates: s = sim * (1 - mask); idx = argmax(s) (first occurrence).
// ---------------------------------------------------------------------------
__global__ void k_greedy(const float* __restrict__ sims,
                         const int* __restrict__ topk,
                         int* __restrict__ sel,
                         float* __restrict__ mask_out) {
    const int b = blockIdx.x;
    const int t = threadIdx.x;
    __shared__ float mask_s[N_];
    __shared__ float rv[32];
    __shared__ int   ri[32];
    __shared__ int   bestIdx;
    mask_s[t] = 0.f;
    __syncthreads();
    if (t < K_) mask_s[topk[b * K_ + t]] = 1.f;
    __syncthreads();
    for (int k = 0; k < K_; ++k) {
        float s = sims[((size_t)b * MPAD + k) * N_ + t] * (1.f - mask_s[t]);
        int id = t;
        red_pair(s, id);
        if ((t & 31) == 0) { rv[t >> 5] = s; ri[t >> 5] = id; }
        __syncthreads();
        if (t < 32) {
            float v2 = rv[t]; int i2 = ri[t];
            red_pair(v2, i2);
            if (t == 0) { bestIdx = i2; sel[b * K_ + k] = i2; }
        }
        __syncthreads();
        if (t == bestIdx) mask_s[t] = 1.f;
        __syncthreads();
    }
    mask_out[b * N_ + t] = mask_s[t];
}

// ---------------------------------------------------------------------------
// Kernel 5a: merge topk+sel and sort ascending (90 elems, insertion sort)
// ---------------------------------------------------------------------------
__global__ void k_sortidx(const int* __restrict__ topk,
                          const int* __restrict__ sel,
                          int* __restrict__ allidx) {
    const int b = blockIdx.x;
    if (threadIdx.x != 0) return;
    int a[2 * K_];
    for (int i = 0; i < K_; ++i) { a[i] = topk[b * K_ + i]; a[K_ + i] = sel[b * K_ + i]; }
    for (int i = 1; i < 2 * K_; ++i) {
        int key = a[i]; int j = i - 1;
        while (j >= 0 && a[j] > key) { a[j + 1] = a[j]; --j; }
        a[j + 1] = key;
    }
    for (int i = 0; i < 2 * K_; ++i) allidx[b * 2 * K_ + i] = a[i];
}

// ---------------------------------------------------------------------------
// Kernel 5b: gather masked_features[b, j, :] = feats[b, allidx[b,j], :]
// grid = B*90 blocks, 192 threads, float4 copies (768 floats / row)
// ---------------------------------------------------------------------------
__global__ void k_gather(const float* __restrict__ feats,
                         const int* __restrict__ allidx,
                         float* __restrict__ out) {
    const int bj = blockIdx.x;
    const int b  = bj / (2 * K_);
    const int row = allidx[bj];
    const float4* src = (const float4*)(feats + ((size_t)b * N_ + row) * D_);
    float4* dst = (float4*)(out + (size_t)bj * D_);
    dst[threadIdx.x] = src[threadIdx.x];
}

// ---------------------------------------------------------------------------
extern "C" void kernel_launch(void* const* d_in, const int* in_sizes, int n_in,
                              void* d_out, int out_size, void* d_ws, size_t ws_size,
                              hipStream_t stream) {
    const float* attn  = (const float*)d_in[0];   // [8,12,1025,1025] f32
    const float* feats = (const float*)d_in[1];   // [8,1024,768] f32
    float* out_mask  = (float*)d_out;             // [8,1024]
    float* out_feats = (float*)d_out + B_ * N_;   // [8,90,768]

    char* ws = (char*)d_ws;
    float* ws_acls =  (float*)(ws);                       //  32768 B
    int*   ws_topk =  (int*)  (ws + 32768);               //   1440 B
    int*   ws_sel  =  (int*)  (ws + 34208);               //   1440 B
    int*   ws_all  =  (int*)  (ws + 35648);               //   2880 B
    float* ws_sims =  (float*)(ws + 40960);               // 8*48*1024*4 B

    k_acls<<<dim3((B_ * N_ + 255) / 256), dim3(256), 0, stream>>>(attn, ws_acls);
    k_topk<<<dim3(B_), dim3(N_), 0, stream>>>(ws_acls, ws_topk);
    k_sims_wmma<<<dim3(B_, 3, 4), dim3(128), 0, stream>>>(feats, ws_topk, ws_sims);
    k_greedy<<<dim3(B_), dim3(N_), 0, stream>>>(ws_sims, ws_topk, ws_sel, out_mask);
    k_sortidx<<<dim3(B_), dim3(32), 0, stream>>>(ws_topk, ws_sel, ws_all);
    k_gather<<<dim3(B_ * 2 * K_), dim3(D_ / 4), 0, stream>>>(feats, ws_all, out_feats);
}